// GATJK_14370960572526
// MI455X (gfx1250) — compile-verified
//
#include <hip/hip_runtime.h>
#include <math.h>

typedef __attribute__((ext_vector_type(2))) float v2f;
typedef __attribute__((ext_vector_type(8))) float v8f;

#ifndef __has_builtin
#define __has_builtin(x) 0
#endif
#if __has_builtin(__builtin_amdgcn_wmma_f32_16x16x4_f32)
#define USE_WMMA_F32 1
#else
#define USE_WMMA_F32 0
#endif

// ---------------------------------------------------------------------------
// GEMM: Y[N,128] = X[N,128] @ W[128,128], fp32 via V_WMMA_F32_16X16X4_F32.
// Block = 256 threads (8 waves), each wave computes a 16x128 output strip.
// W staged in LDS (64KB of the 320KB WGP pool).
// ---------------------------------------------------------------------------
__global__ __launch_bounds__(256) void gemm128_wmma(
    const float* __restrict__ X, const float* __restrict__ W,
    float* __restrict__ Y, int Nrows)
{
    __shared__ float Wl[128 * 128];
    for (int i = threadIdx.x; i < 128 * 128; i += 256) Wl[i] = W[i];
    __syncthreads();

#if USE_WMMA_F32
    const int lane  = threadIdx.x & 31;
    const int wave  = threadIdx.x >> 5;
    const int nlane = lane & 15;
    const int hi    = lane >> 4;      // 0: K=0,1 half; 1: K=2,3 half
    const int khalf = hi * 2;
    const int row0  = blockIdx.x * 128 + wave * 16;

    int arow = row0 + nlane;
    if (arow >= Nrows) arow = Nrows - 1;   // clamp: OOB rows never stored
    const float* __restrict__ xrow = X + (size_t)arow * 128;

    v8f acc[8];
    v8f zero = {0.f, 0.f, 0.f, 0.f, 0.f, 0.f, 0.f, 0.f};
#pragma unroll
    for (int t = 0; t < 8; ++t) acc[t] = zero;

    for (int k = 0; k < 128; k += 4) {
        v2f a;
        a.x = xrow[k + khalf];
        a.y = xrow[k + khalf + 1];
#pragma unroll
        for (int t = 0; t < 8; ++t) {
            v2f b;
            b.x = Wl[(k + khalf)     * 128 + t * 16 + nlane];
            b.y = Wl[(k + khalf + 1) * 128 + t * 16 + nlane];
            acc[t] = __builtin_amdgcn_wmma_f32_16x16x4_f32(
                false, a, false, b, (short)0, acc[t], false, false);
        }
    }

    const int mbase = hi * 8;
#pragma unroll
    for (int t = 0; t < 8; ++t) {
#pragma unroll
        for (int v = 0; v < 8; ++v) {
            int r = row0 + mbase + v;
            if (r < Nrows)
                Y[(size_t)r * 128 + t * 16 + nlane] = acc[t][v];
        }
    }
#else
    // Scalar fallback (should not be taken on gfx1250 toolchains)
    int r = blockIdx.x * 128 + (int)(threadIdx.x >> 1);
    int cbase = (threadIdx.x & 1) * 64;
    if (r < Nrows) {
        const float* xr = X + (size_t)r * 128;
        for (int c = 0; c < 64; ++c) {
            float acc = 0.f;
            for (int k = 0; k < 128; ++k) acc += xr[k] * Wl[k * 128 + cbase + c];
            Y[(size_t)r * 128 + cbase + c] = acc;
        }
    }
#endif
}

// ---------------------------------------------------------------------------
// Per-node attention dots: asrc[n,h] = <H[n,h,:], att_src[h,:]> (C=64, H=2)
// One wave per node; lanes 0-15 -> head 0, lanes 16-31 -> head 1.
// ---------------------------------------------------------------------------
__global__ void att_dots_kernel(const float* __restrict__ Hb,
                                const float* __restrict__ att_s,
                                const float* __restrict__ att_d,
                                float* __restrict__ asrc,
                                float* __restrict__ adst, int N)
{
    int gid  = blockIdx.x * blockDim.x + threadIdx.x;
    int node = gid >> 5;
    int lane = threadIdx.x & 31;
    if (node >= N) return;
    float4 h4 = ((const float4*)(Hb + (size_t)node * 128))[lane];
    float4 s4 = ((const float4*)att_s)[lane];
    float4 d4 = ((const float4*)att_d)[lane];
    float ps = h4.x * s4.x + h4.y * s4.y + h4.z * s4.z + h4.w * s4.w;
    float pd = h4.x * d4.x + h4.y * d4.y + h4.z * d4.z + h4.w * d4.w;
    for (int off = 1; off < 16; off <<= 1) {
        ps += __shfl_xor(ps, off, 32);
        pd += __shfl_xor(pd, off, 32);
    }
    if ((lane & 15) == 0) {
        int head = lane >> 4;
        asrc[node * 2 + head] = ps;
        adst[node * 2 + head] = pd;
    }
}

__global__ void fill_f32(float* __restrict__ p, float v, long n) {
    long i = (long)blockIdx.x * blockDim.x + threadIdx.x;
    if (i < n) p[i] = v;
}

__device__ __forceinline__ void atomicMaxF(float* addr, float val) {
    // monotone bit-trick: non-negative -> signed int max; negative -> unsigned min
    if (val >= 0.0f) atomicMax((int*)addr, __float_as_int(val));
    else             atomicMin((unsigned int*)addr, __float_as_uint(val));
}

__device__ __forceinline__ void edge_nodes(const long long* ei, int E, int e,
                                           long long& s, long long& d) {
    if (e < E) { s = ei[e]; d = ei[E + e]; }
    else       { s = d = (long long)(e - E); }  // self loops appended
}

// thread per (edge, head): segment max of leaky_relu(a_src[s]+a_dst[d])
__global__ void edge_max_kernel(const long long* __restrict__ ei, int E, int E2,
                                const float* __restrict__ asrc,
                                const float* __restrict__ adst,
                                float* __restrict__ m)
{
    int idx = blockIdx.x * blockDim.x + threadIdx.x;
    if (idx >= E2 * 2) return;
    int e = idx >> 1, head = idx & 1;
    long long s, d; edge_nodes(ei, E, e, s, d);
    float v = asrc[s * 2 + head] + adst[d * 2 + head];
    v = v > 0.0f ? v : 0.2f * v;
    atomicMaxF(&m[d * 2 + head], v);
}

// thread per (edge, head): ex = exp(e - m[dst]); store; segment-sum denom
__global__ void edge_exp_kernel(const long long* __restrict__ ei, int E, int E2,
                                const float* __restrict__ asrc,
                                const float* __restrict__ adst,
                                const float* __restrict__ m,
                                float* __restrict__ exv,
                                float* __restrict__ denom)
{
    int idx = blockIdx.x * blockDim.x + threadIdx.x;
    if (idx >= E2 * 2) return;
    int e = idx >> 1, head = idx & 1;
    long long s, d; edge_nodes(ei, E, e, s, d);
    float v = asrc[s * 2 + head] + adst[d * 2 + head];
    v = v > 0.0f ? v : 0.2f * v;
    float ex = __expf(v - m[d * 2 + head]);
    exv[idx] = ex;
    atomicAdd(&denom[d * 2 + head], ex);
}

// one wave per edge: agg[dst] += alpha * h[src]  (float4 gather, f32 atomics)
__global__ void scatter_kernel(const long long* __restrict__ ei, int E, int E2,
                               const float* __restrict__ Hb,
                               const float* __restrict__ exv,
                               const float* __restrict__ denom,
                               float* __restrict__ agg)
{
    int gid  = blockIdx.x * blockDim.x + threadIdx.x;
    int e    = gid >> 5;
    int lane = threadIdx.x & 31;
    if (e >= E2) return;
    long long s, d; edge_nodes(ei, E, e, s, d);
    int head = lane >> 4;
    float alpha = exv[(size_t)e * 2 + head] / (denom[d * 2 + head] + 1e-16f);
    float4 h4 = ((const float4*)(Hb + (size_t)s * 128))[lane];
    float* op = agg + (size_t)d * 128 + lane * 4;
    atomicAdd(op + 0, alpha * h4.x);
    atomicAdd(op + 1, alpha * h4.y);
    atomicAdd(op + 2, alpha * h4.z);
    atomicAdd(op + 3, alpha * h4.w);
}

// x1 = elu( BN( agg + b1 ) )
__global__ void bn_elu_kernel(const float* __restrict__ agg, const float* __restrict__ b,
                              const float* __restrict__ gamma, const float* __restrict__ beta,
                              const float* __restrict__ mean, const float* __restrict__ var,
                              float* __restrict__ x1, long n)
{
    long i = (long)blockIdx.x * blockDim.x + threadIdx.x;
    if (i >= n) return;
    int f = (int)(i & 127);
    float v = agg[i] + b[f];
    v = (v - mean[f]) * rsqrtf(var[f] + 1e-5f) * gamma[f] + beta[f];
    x1[i] = v > 0.0f ? v : (__expf(v) - 1.0f);
}

__global__ void bias_kernel(const float* __restrict__ agg, const float* __restrict__ b,
                            float* __restrict__ x2, long n)
{
    long i = (long)blockIdx.x * blockDim.x + threadIdx.x;
    if (i >= n) return;
    x2[i] = agg[i] + b[(int)(i & 127)];
}

// JK max + classifier (128->40) + log_softmax. One 64-thread block per node.
__global__ __launch_bounds__(64) void final_kernel(
    const float* __restrict__ x1, const float* __restrict__ x2,
    const float* __restrict__ Wf, const float* __restrict__ bf,
    float* __restrict__ out, int N)
{
    __shared__ float xj[128];
    __shared__ float logits[40];
    __shared__ float red[2];
    int n = blockIdx.x;
    int t = threadIdx.x;
    for (int i = t; i < 128; i += 64)
        xj[i] = fmaxf(x1[(size_t)n * 128 + i], x2[(size_t)n * 128 + i]);
    __syncthreads();
    if (t < 40) {
        float acc = bf[t];
        for (int k = 0; k < 128; ++k) acc += xj[k] * Wf[k * 40 + t];
        logits[t] = acc;
    }
    __syncthreads();
    if (t == 0) {
        float mx = logits[0];
        for (int i = 1; i < 40; ++i) mx = fmaxf(mx, logits[i]);
        float sum = 0.f;
        for (int i = 0; i < 40; ++i) sum += __expf(logits[i] - mx);
        red[0] = mx;
        red[1] = __logf(sum);
    }
    __syncthreads();
    if (t < 40) out[(size_t)n * 40 + t] = logits[t] - red[0] - red[1];
}

// ---------------------------------------------------------------------------
extern "C" void kernel_launch(void* const* d_in, const int* in_sizes, int n_in,
                              void* d_out, int out_size, void* d_ws, size_t ws_size,
                              hipStream_t stream)
{
    const float*     x   = (const float*)d_in[0];
    const long long* ei  = (const long long*)d_in[1];   // int64 edge_index [2,E]
    const float*     W1  = (const float*)d_in[2];
    const float*     as1 = (const float*)d_in[3];
    const float*     ad1 = (const float*)d_in[4];
    const float*     b1  = (const float*)d_in[5];
    const float*     g   = (const float*)d_in[6];
    const float*     bta = (const float*)d_in[7];
    const float*     mu  = (const float*)d_in[8];
    const float*     var = (const float*)d_in[9];
    const float*     W2  = (const float*)d_in[10];
    const float*     as2 = (const float*)d_in[11];
    const float*     ad2 = (const float*)d_in[12];
    const float*     b2  = (const float*)d_in[13];
    const float*     Wf  = (const float*)d_in[14];
    const float*     bf  = (const float*)d_in[15];
    float* out = (float*)d_out;
    (void)n_in; (void)out_size; (void)ws_size;

    const int  N  = in_sizes[0] / 128;
    const int  E  = in_sizes[1] / 2;
    const int  E2 = E + N;             // + self loops
    const long NF = (long)N * 128;

    float* ws   = (float*)d_ws;
    float* bufH = ws;  ws += NF;       // GEMM output h (per layer)
    float* agg  = ws;  ws += NF;       // aggregation accumulator
    float* x1   = ws;  ws += NF;
    float* x2   = ws;  ws += NF;
    float* asrc = ws;  ws += (long)N * 2;
    float* adst = ws;  ws += (long)N * 2;
    float* mbuf = ws;  ws += (long)N * 2;
    float* den  = ws;  ws += (long)N * 2;
    float* exv  = ws;  ws += (long)E2 * 2;

    dim3 blk(256);
    int gGemm     = (N + 127) / 128;
    int gNodeWave = (int)(((long)N * 32 + 255) / 256);
    int gEdge2    = (int)(((long)E2 * 2 + 255) / 256);
    int gEdgeWave = (int)(((long)E2 * 32 + 255) / 256);
    int gNF       = (int)((NF + 255) / 256);
    int gN2       = (int)(((long)N * 2 + 255) / 256);

    for (int layer = 0; layer < 2; ++layer) {
        const float* Xin  = (layer == 0) ? x   : x1;
        const float* Wmat = (layer == 0) ? W1  : W2;
        const float* asl  = (layer == 0) ? as1 : as2;
        const float* adl  = (layer == 0) ? ad1 : ad2;

        gemm128_wmma<<<gGemm, blk, 0, stream>>>(Xin, Wmat, bufH, N);
        att_dots_kernel<<<gNodeWave, blk, 0, stream>>>(bufH, asl, adl, asrc, adst, N);
        fill_f32<<<gN2, blk, 0, stream>>>(mbuf, -INFINITY, (long)N * 2);
        fill_f32<<<gN2, blk, 0, stream>>>(den, 0.0f, (long)N * 2);
        fill_f32<<<gNF, blk, 0, stream>>>(agg, 0.0f, NF);
        edge_max_kernel<<<gEdge2, blk, 0, stream>>>(ei, E, E2, asrc, adst, mbuf);
        edge_exp_kernel<<<gEdge2, blk, 0, stream>>>(ei, E, E2, asrc, adst, mbuf, exv, den);
        scatter_kernel<<<gEdgeWave, blk, 0, stream>>>(ei, E, E2, bufH, exv, den, agg);
        if (layer == 0)
            bn_elu_kernel<<<gNF, blk, 0, stream>>>(agg, b1, g, bta, mu, var, x1, NF);
        else
            bias_kernel<<<gNF, blk, 0, stream>>>(agg, b2, x2, NF);
    }

    final_kernel<<<N, dim3(64), 0, stream>>>(x1, x2, Wf, bf, out, N);
}